// TnoMatrix_61916248539624
// MI455X (gfx1250) — compile-verified
//
#include <hip/hip_runtime.h>

// Problem constants (reference: B=2, N=2048, D=64, CAUSAL=False)
#define BB 2
#define NN 2048
#define DD 64
#define LW 4096                         // band length per copy
#define PADX 256
#define LXP (NN + 2*PADX)               // 2560
#define LXS (LXP + (LXP/8)*2)           // 3200 skewed floats in LDS
#define WP_ELEMS (DD * LW)              // 262144 floats per band copy
#define XP_ELEMS (BB * DD * LXP)        // 327680 floats
#define KMACRO 143                      // macro-steps of 16 K (4 WMMAs each)

typedef __attribute__((ext_vector_type(2))) float v2f;
typedef __attribute__((ext_vector_type(8))) float v8f;

// ---------------------------------------------------------------------------
// Toeplitz band value: w_d[j]; w[0]=t[0,d], w[j>0]=t[2N-j,d], w[j<0]=t[-j,d]
// ---------------------------------------------------------------------------
__device__ __forceinline__ float wband(const float* __restrict__ t, int d, int j) {
    if (j == 0)                return t[d];
    if (j > 0  && j <  NN)     return t[(2 * NN - j) * DD + d];
    if (j < 0  && j > -NN)     return t[(-j) * DD + d];
    return 0.0f;
}

// Prep 1: two shifted band copies. Wp[d][p]=w[p-2047], Wp2[d][p]=w[p-2046].
__global__ void tno_build_w(const float* __restrict__ t,
                            float* __restrict__ Wp, float* __restrict__ Wp2) {
    int idx = blockIdx.x * blockDim.x + threadIdx.x;   // D*LW threads
    int d = idx >> 12;
    int p = idx & (LW - 1);
    Wp [idx] = wband(t, d, p - (NN - 1));
    Wp2[idx] = wband(t, d, p - (NN - 2));
}

// Prep 2: transpose+pad  Xp[b][d][q] = x[b, q-PADX, d] (zero outside [0,N))
__global__ void tno_build_x(const float* __restrict__ x, float* __restrict__ Xp) {
    int idx = blockIdx.x * blockDim.x + threadIdx.x;   // B*D*LXP threads
    int b = idx / (DD * LXP);
    int rem = idx - b * (DD * LXP);
    int d = rem / LXP;
    int q = rem - d * LXP;
    int pos = q - PADX;
    float v = 0.0f;
    if (pos >= 0 && pos < NN) v = x[((size_t)b * NN + pos) * DD + d];
    Xp[idx] = v;
}

// CDNA5 async copy: 16 bytes global -> LDS, tracked by ASYNCcnt (GV mode).
__device__ __forceinline__ void async_b128(unsigned lds_byte, const float* src) {
    asm volatile("global_load_async_to_lds_b128 %0, %1, off"
                 :: "v"(lds_byte), "v"(src) : "memory");
}

// ---------------------------------------------------------------------------
// Main: workgroup = (b, d); wave w = row-group i0 = 16w (256 output rows).
// y[16*(i0+i)+r] = sum_u w_d[u-r] * x[b, u+16*(i0+i), d]
// A[r][K] = w(u0+K-r) from dual-copy LDS band (8B aligned per lane parity),
// B[K][i] = x(u0+K+16*(i0+i)) from skew-8 LDS (conflict-free, 8B aligned).
// Per 16-K macro-step: A offsets {0,4,8,12} floats (base += 16),
//                      B offsets {0,4,10,14} floats (base += 20, exact since
//                      (q+16m)>>3 == (q>>3)+2m for the skew idx q+2*(q>>3)).
// ---------------------------------------------------------------------------
__global__ __launch_bounds__(256) void tno_wmma_main(const float* __restrict__ Wp,
                                                     const float* __restrict__ Wp2,
                                                     const float* __restrict__ Xp,
                                                     float* __restrict__ out) {
    __shared__ __align__(16) float lds_w1[LW];    // w[q-2047]
    __shared__ __align__(16) float lds_w2[LW];    // w[q-2046]
    __shared__ __align__(16) float lds_x[LXS];    // skew: idx = q + 2*(q>>3)

    const int tid = threadIdx.x;
    const int b = blockIdx.x >> 6;
    const int d = blockIdx.x & 63;

    const float* wrow1 = Wp  + (size_t)d * LW;
    const float* wrow2 = Wp2 + (size_t)d * LW;
    const float* xrow  = Xp  + (size_t)(b * DD + d) * LXP;

    const unsigned w1b = (unsigned)(size_t)&lds_w1[0];
    const unsigned w2b = (unsigned)(size_t)&lds_w2[0];
    const unsigned xb  = (unsigned)(size_t)&lds_x[0];

    // Fill LDS: band copies (1024 16B-chunks each), skewed x (640 chunks).
    for (int c = tid; c < LW / 4; c += 256) {
        async_b128(w1b + 16u * c, wrow1 + 4 * c);
        async_b128(w2b + 16u * c, wrow2 + 4 * c);
    }
    for (int c = tid; c < LXP / 4; c += 256) {
        async_b128(xb + 16u * c + 8u * (c >> 1), xrow + 4 * c);  // skew-8 dst
    }
    asm volatile("s_wait_asynccnt 0" ::: "memory");
    __syncthreads();

    const int lane = tid & 31;
    const int wave = tid >> 5;
    const int i0   = wave << 4;
    const int r    = lane & 15;                 // A row / B-col index
    const int gsel = lane >> 4;                 // K half select
    const int u_lo = -16 * (i0 + 15);

    // A base: copy chosen by r parity so every float2 is 8B aligned.
    const float* wbase = (r & 1) ? (lds_w1 + (NN - 1)) : (lds_w2 + (NN - 2));
    const float* ap = wbase + (u_lo + 2 * gsel - r);
    const int q0 = u_lo + 2 * gsel + 16 * (i0 + r) + PADX;   // >= 16, even
    const float* bp = lds_x + (q0 + 2 * (q0 >> 3));          // skewed base

    v8f acc = {0.f, 0.f, 0.f, 0.f, 0.f, 0.f, 0.f, 0.f};

    for (int m = 0; m < KMACRO; ++m) {
        v2f a0 = *(const v2f*)(ap + 0);
        v2f a1 = *(const v2f*)(ap + 4);
        v2f a2 = *(const v2f*)(ap + 8);
        v2f a3 = *(const v2f*)(ap + 12);
        v2f b0 = *(const v2f*)(bp + 0);
        v2f b1 = *(const v2f*)(bp + 4);
        v2f b2 = *(const v2f*)(bp + 10);
        v2f b3 = *(const v2f*)(bp + 14);
        acc = __builtin_amdgcn_wmma_f32_16x16x4_f32(false, a0, false, b0, (short)0, acc, false, false);
        acc = __builtin_amdgcn_wmma_f32_16x16x4_f32(false, a1, false, b1, (short)0, acc, false, false);
        acc = __builtin_amdgcn_wmma_f32_16x16x4_f32(false, a2, false, b2, (short)0, acc, false, false);
        acc = __builtin_amdgcn_wmma_f32_16x16x4_f32(false, a3, false, b3, (short)0, acc, false, false);
        ap += 16;
        bp += 20;
    }

    // acc[v] -> out[b, 16*(i0 + r) + v + 8*gsel, d]
    const int nbase = 16 * (i0 + r) + 8 * gsel;
    float* op = out + ((size_t)b * NN + nbase) * DD + d;
#pragma unroll
    for (int v = 0; v < 8; ++v) op[(size_t)v * DD] = acc[v];
}

// ---------------------------------------------------------------------------
extern "C" void kernel_launch(void* const* d_in, const int* in_sizes, int n_in,
                              void* d_out, int out_size, void* d_ws, size_t ws_size,
                              hipStream_t stream) {
    const float* x = (const float*)d_in[0];   // (B, N, D) f32
    const float* t = (const float*)d_in[1];   // (2N, D) f32
    float* out = (float*)d_out;               // (B, N, D) f32

    float* Wp  = (float*)d_ws;
    float* Wp2 = Wp + WP_ELEMS;
    float* Xp  = Wp2 + WP_ELEMS;

    tno_build_w<<<WP_ELEMS / 256, 256, 0, stream>>>(t, Wp, Wp2);
    tno_build_x<<<XP_ELEMS / 256, 256, 0, stream>>>(x, Xp);
    tno_wmma_main<<<BB * DD, 256, 0, stream>>>(Wp, Wp2, Xp, out);
}